// GraphConvolution_26053271617787
// MI455X (gfx1250) — compile-verified
//
#include <hip/hip_runtime.h>
#include <hip/hip_bf16.h>
#include <stddef.h>

// ---------------------------------------------------------------------------
// GCN layer for MI455X (gfx1250, wave32, WMMA):
//   x   = (features * drop_mask) @ W          [50000 x 128]  (f32 WMMA 16x16x4)
//   agg = segment_sum(vals * x[cols], rows)   (L2-resident gather + hw f32 atomics)
//   out = relu(agg + b)
//
// Roofline: GEMM moves ~230MB -> ~10us at 23.3TB/s; 6.6 GFLOP is trivial for
// WMMA, so full-precision V_WMMA_F32_16X16X4_F32 is the right choice (exact
// f32 semantics, still bandwidth-bound).  The SpMM gather operand x is 25.6MB
// and lives in the 192MB L2; the scatter uses non-returning hardware
// global_atomic_add_f32 at the L2 atomic units.
// ---------------------------------------------------------------------------

#define GCN_N     50000
#define GCN_FIN   512
#define GCN_FOUT  128
#define LDS_PITCH 516   // 512 + 4 floats pad -> conflict-free column reads

typedef float v2f __attribute__((ext_vector_type(2)));
typedef float v8f __attribute__((ext_vector_type(8)));

// Guaranteed hardware f32 atomic add (global_atomic_add_f32, non-returning ->
// STOREcnt, fire-and-forget).  Plain atomicAdd(float*) may CAS-expand.
__device__ __forceinline__ void hw_atomic_add_f32(float* p, float v)
{
#if defined(__has_builtin) && __has_builtin(__builtin_amdgcn_global_atomic_fadd_f32)
    (void)unsafeAtomicAdd(p, v);
#else
    atomicAdd(p, v);
#endif
}

// ---------------------------------------------------------------------------
// Kernel 1: fused dropout + GEMM via V_WMMA_F32_16X16X4_F32
//   grid.x = N/16 (=3125 row tiles), block = 256 threads = 8 waves
//   each block stages one 16x512 A-tile in LDS; wave w computes the 16x16
//   output tile at columns [16w, 16w+16).
// ---------------------------------------------------------------------------
__global__ __launch_bounds__(256)
void gcn_gemm_wmma(const float* __restrict__ A,   // features [N, 512]
                   const float* __restrict__ Mk,  // drop mask [N, 512]
                   const float* __restrict__ W,   // [512, 128] row-major
                   float* __restrict__ X)         // [N, 128]
{
    __shared__ float As[16 * LDS_PITCH];

    const int row0 = blockIdx.x * 16;
    const int tid  = threadIdx.x;

    // ---- stage A-tile (dropout applied) into LDS: 2048 float4s / 256 thr ----
    #pragma unroll
    for (int i = 0; i < 8; ++i) {
        const int idx = tid + i * 256;      // float4 index in [0, 2048)
        const int r   = idx >> 7;           // row within tile
        const int c   = (idx & 127) << 2;   // column (float)
        const size_t g = (size_t)(row0 + r) * GCN_FIN + c;
        const float4 a = *(const float4*)(A  + g);
        const float4 m = *(const float4*)(Mk + g);
        float4 v;
        v.x = a.x * m.x; v.y = a.y * m.y; v.z = a.z * m.z; v.w = a.w * m.w;
        *(float4*)(&As[r * LDS_PITCH + c]) = v;
    }
    __syncthreads();

    const int wave  = tid >> 5;
    const int lane  = tid & 31;
    const int n0    = wave << 4;            // this wave's F_OUT column base
    const int mrow  = lane & 15;            // A row / B column index
    const int khalf = (lane >> 4) << 1;     // lanes 16..31 hold K+2, K+3
    const int ncol  = n0 + mrow;            // global B/N column for this lane

    v8f acc = {0.f, 0.f, 0.f, 0.f, 0.f, 0.f, 0.f, 0.f};

    #pragma unroll 4
    for (int k = 0; k < GCN_FIN; k += 4) {
        const int kb = k + khalf;
        // A fragment: row mrow, K = kb, kb+1 (8B-aligned ds_load_b64)
        const v2f afrag = *(const v2f*)(&As[mrow * LDS_PITCH + kb]);
        // B fragment: W[kb, ncol], W[kb+1, ncol] (column ncol, L2-hot)
        v2f bfrag;
        bfrag.x = W[(size_t)kb       * GCN_FOUT + ncol];
        bfrag.y = W[(size_t)(kb + 1) * GCN_FOUT + ncol];
        // prefetch W a couple of K-steps ahead (global_prefetch_b8)
        __builtin_prefetch(&W[(size_t)(kb + 8) * GCN_FOUT + ncol], 0, 1);
        // D = A(16x4,f32) * B(4x16,f32) + C   (full f32 precision)
        acc = __builtin_amdgcn_wmma_f32_16x16x4_f32(
            /*neg_a=*/false, afrag, /*neg_b=*/false, bfrag,
            /*c_mod=*/(short)0, acc, /*reuse_a=*/false, /*reuse_b=*/false);
    }

    // ---- store C/D tile: VGPR i -> row (i + 8*(lane>=16)), col = lane&15 ----
    const int mbase = (lane >> 4) << 3;
    float* Xt = X + (size_t)row0 * GCN_FOUT + n0;
    #pragma unroll
    for (int i = 0; i < 8; ++i) {
        Xt[(size_t)(mbase + i) * GCN_FOUT + (lane & 15)] = acc[i];
    }
}

// ---------------------------------------------------------------------------
// Kernel 0: zero the aggregation buffer (needed every call; ws not re-zeroed)
// ---------------------------------------------------------------------------
__global__ void gcn_zero(float4* __restrict__ p, int n4)
{
    int i = blockIdx.x * blockDim.x + threadIdx.x;
    if (i < n4) p[i] = make_float4(0.f, 0.f, 0.f, 0.f);
}

// ---------------------------------------------------------------------------
// Kernel 2: edge-parallel SpMM scatter.  One wave per edge; lane q owns
// float4 #q of the 128-wide row.  x fits in the 192MB L2, so the gather is
// L2-resident; scatter uses non-returning hardware f32 atomics.
// ---------------------------------------------------------------------------
__global__ __launch_bounds__(256)
void gcn_spmm_scatter(const float* __restrict__ x,
                      const int* __restrict__ rows,
                      const int* __restrict__ cols,
                      const float* __restrict__ vals,
                      float* __restrict__ agg,
                      int E)
{
    const int gid = blockIdx.x * blockDim.x + threadIdx.x;
    const int e   = gid >> 5;
    if (e >= E) return;
    const int q   = gid & 31;

    const int   row = rows[e];   // uniform across the wave -> scalarized
    const int   col = cols[e];
    const float v   = vals[e];

    const float4 xv = *(const float4*)(x + (size_t)col * GCN_FOUT + (q << 2));
    float* dst = agg + (size_t)row * GCN_FOUT + (q << 2);
    hw_atomic_add_f32(dst + 0, v * xv.x);
    hw_atomic_add_f32(dst + 1, v * xv.y);
    hw_atomic_add_f32(dst + 2, v * xv.z);
    hw_atomic_add_f32(dst + 3, v * xv.w);
}

// ---------------------------------------------------------------------------
// Kernel 3: fused bias + ReLU epilogue
// ---------------------------------------------------------------------------
__global__ void gcn_epilogue(const float* __restrict__ agg,
                             const float* __restrict__ b,
                             float* __restrict__ out,
                             int n)
{
    int i = blockIdx.x * blockDim.x + threadIdx.x;
    if (i < n) {
        float v = agg[i] + b[i & (GCN_FOUT - 1)];
        out[i] = v > 0.f ? v : 0.f;
    }
}

// ---------------------------------------------------------------------------
extern "C" void kernel_launch(void* const* d_in, const int* in_sizes, int n_in,
                              void* d_out, int out_size, void* d_ws, size_t ws_size,
                              hipStream_t stream)
{
    const float* features = (const float*)d_in[0];
    const float* dropmask = (const float*)d_in[1];
    const float* W        = (const float*)d_in[2];
    const float* bias     = (const float*)d_in[3];
    const int*   rows     = (const int*)d_in[4];
    const int*   cols     = (const int*)d_in[5];
    const float* vals     = (const float*)d_in[6];
    const int    E        = in_sizes[4];

    float* x   = (float*)d_ws;                       // [N, 128]
    float* agg = x + (size_t)GCN_N * GCN_FOUT;       // [N, 128]
    float* out = (float*)d_out;

    const int total = GCN_N * GCN_FOUT;              // 6.4M

    // zero agg (atomics accumulate into it)
    gcn_zero<<<(total / 4 + 255) / 256, 256, 0, stream>>>((float4*)agg, total / 4);

    // dense branch: dropout + GEMM on WMMA
    gcn_gemm_wmma<<<GCN_N / 16, 256, 0, stream>>>(features, dropmask, W, x);

    // sparse scatter: one wave per edge
    const long long sp_threads = (long long)E * 32;
    gcn_spmm_scatter<<<(int)((sp_threads + 255) / 256), 256, 0, stream>>>(
        x, rows, cols, vals, agg, E);

    // bias + relu
    gcn_epilogue<<<(total + 255) / 256, 256, 0, stream>>>(agg, bias, out, total);
}